// MaterialModelFinder_43413529428309
// MI455X (gfx1250) — compile-verified
//
#include <hip/hip_runtime.h>

// MI455X / gfx1250, wave32. FEM equilibrium-residual loss.
//
// Strategy: fold time-invariant geometry into per-element 6x6 stiffness K_e,
// then F_e[6,64] = K_e * U_e[6,64] via V_WMMA_F32_16X16X4_F32 (one wave per
// element, 2 K-chunks x 4 time-tiles = 8 WMMAs). Scatter with f32 global
// atomics into a node-major [N,2,T] accumulator that matches reaction_hist's
// layout, then a coalesced float4 masked sum-of-squares reduction.
// Problem is HBM/atomic bound (~300MB traffic, ~1 GFLOP) -> f32 end to end.
// All scratch offsets fit in u32 (51MB arrays) -> 32-bit address math, with
// the time-tile index folded into the instruction's immediate offset.

typedef float v8f __attribute__((ext_vector_type(8)));
typedef float v2f __attribute__((ext_vector_type(2)));

#define FEM_EPS 1e-8f

// ---------------------------------------------------------------- zero fill
__global__ void fem_zero_f32v4(float4* __restrict__ p, long n4) {
  long i = (long)blockIdx.x * blockDim.x + threadIdx.x;
  long stride = (long)gridDim.x * blockDim.x;
  const float4 z = make_float4(0.f, 0.f, 0.f, 0.f);
  for (; i < n4; i += stride) p[i] = z;
}

__global__ void fem_zero_f32(float* __restrict__ p, int n) {
  int i = blockIdx.x * blockDim.x + threadIdx.x;
  if (i < n) p[i] = 0.0f;
}

// ------------------------------------------------- [T, N*2] -> [N*2, T]
// blockIdx.y = t (no divisions). Reads coalesced; scattered writes are
// absorbed by the 192MB L2 (whole array = 51MB).
__global__ void fem_transpose_t_major_to_n_major(const float* __restrict__ src,
                                                 float* __restrict__ dst,
                                                 int n2, int T) {
  const int t = blockIdx.y;
  int nc = blockIdx.x * blockDim.x + threadIdx.x;
  if (nc < n2)
    dst[(unsigned)nc * (unsigned)T + (unsigned)t] =
        src[(unsigned)t * (unsigned)n2 + (unsigned)nc];
}

// ------------------------------------------------------------ element kernel
// One wave (32 lanes) per element. T == 64 (4 time-tiles of 16).
__device__ __forceinline__ float fem_sel3(int i, float v0, float v1, float v2) {
  return (i == 0) ? v0 : ((i == 1) ? v1 : v2);
}

__global__ void fem_elem_forces_wmma(const float* __restrict__ coords,
                                     const float* __restrict__ u_tr,  // [N*2, T]
                                     const float* __restrict__ lam_,
                                     const float* __restrict__ mu_,
                                     const int*   __restrict__ conn,  // [E,3]
                                     float*       __restrict__ f_acc, // [N*2, T]
                                     int n_elem, int T) {
  int wave = (int)((blockIdx.x * blockDim.x + threadIdx.x) >> 5);
  int lane = threadIdx.x & 31;
  if (wave >= n_elem) return;  // wave-uniform guard: EXEC all-1 inside
  const int e = wave;

  // --- geometry (computed redundantly per lane; cheap, cached loads) ---
  const int na = conn[3 * e + 0];
  const int nb = conn[3 * e + 1];
  const int nc = conn[3 * e + 2];
  const float x0 = coords[2 * na + 0], y0 = coords[2 * na + 1];
  const float x1 = coords[2 * nb + 0], y1 = coords[2 * nb + 1];
  const float x2 = coords[2 * nc + 0], y2 = coords[2 * nc + 1];
  const float b0 = y1 - y2, b1 = y2 - y0, b2 = y0 - y1;
  const float c0 = x2 - x1, c1 = x0 - x2, c2 = x1 - x0;
  const float twoA = x0 * b0 + x1 * b1 + x2 * b2;
  const float tas = (fabsf(twoA) < FEM_EPS) ? FEM_EPS : twoA;
  const float s = 0.5f * fabsf(twoA) / (tas * tas);  // area / twoA_safe^2
  const float lam = lam_[e];
  const float mu  = mu_[e];
  const float lp2 = lam + 2.0f * mu;

  // K_e[p][q] = s * (Bhat col p)^T D (Bhat col q), p,q in [0,6)
  auto ke = [&](int p, int q) -> float {
    if (p >= 6) return 0.0f;  // padded rows of the 16x4 A tile
    const float bi = fem_sel3(p >> 1, b0, b1, b2);
    const float ci = fem_sel3(p >> 1, c0, c1, c2);
    const float bj = fem_sel3(q >> 1, b0, b1, b2);
    const float cj = fem_sel3(q >> 1, c0, c1, c2);
    const bool pe = (p & 1) == 0, qe = (q & 1) == 0;
    float v;
    if (pe && qe)        v = bi * lp2 * bj + ci * mu * cj;
    else if (pe && !qe)  v = bi * lam * cj + ci * mu * bj;
    else if (!pe && qe)  v = ci * lam * bj + bi * mu * cj;
    else                 v = ci * lp2 * cj + bi * mu * bj;
    return s * v;
  };

  const bool hi = lane >= 16;
  const int  M  = lane & 15;  // A row / B column(time) / D column within tile

  // A fragments (16x4 f32): VGPR0 = {K0 | K2}, VGPR1 = {K1 | K3} per ISA 7.12.2
  v2f a0, a1;
  a0.x = ke(M, hi ? 2 : 0);
  a0.y = ke(M, hi ? 3 : 1);
  a1.x = hi ? 0.0f : ke(M, 4);   // K-chunk 1: cols 4,5 live; cols 6,7 == 0
  a1.y = hi ? 0.0f : ke(M, 5);

  // Per-DOF 32-bit element offsets into [N*2, T] arrays, with this lane's M
  // folded in once. All arrays are 12.8M elements -> offsets fit u32 and the
  // time-tile (tb = 0,16,32,48) becomes an immediate offset in the encoding.
  const unsigned uT = (unsigned)T, uM = (unsigned)M;
  const unsigned off0 = ((unsigned)na * 2u + 0u) * uT + uM;
  const unsigned off1 = ((unsigned)na * 2u + 1u) * uT + uM;
  const unsigned off2 = ((unsigned)nb * 2u + 0u) * uT + uM;
  const unsigned off3 = ((unsigned)nb * 2u + 1u) * uT + uM;
  const unsigned off4 = ((unsigned)nc * 2u + 0u) * uT + uM;
  const unsigned off5 = ((unsigned)nc * 2u + 1u) * uT + uM;
  const unsigned offX = hi ? off2 : off0;  // bf0.x source
  const unsigned offY = hi ? off3 : off1;  // bf0.y source

#pragma unroll
  for (int tb = 0; tb < 64; tb += 16) {
    const unsigned utb = (unsigned)tb;
    // B fragments (4x16 f32), N = time: VGPR0 = {K0 | K2}, VGPR1 = {K1 | K3}.
    // Chunk-1 hi lanes feed B rows K=6,7 whose A columns are all zero, so
    // their loaded values are irrelevant -> unconditional loads, no EXEC
    // branching around VMEM.
    v2f bf0, bf1;
    bf0.x = u_tr[offX + utb];
    bf0.y = u_tr[offY + utb];
    bf1.x = u_tr[off4 + utb];
    bf1.y = u_tr[off5 + utb];

    v8f acc = {};
    // D(16x16) += A(16x4) * B(4x16); two K-chunks cover the 6-wide contraction
    acc = __builtin_amdgcn_wmma_f32_16x16x4_f32(false, a0, false, bf0,
                                                (short)0, acc, false, false);
    acc = __builtin_amdgcn_wmma_f32_16x16x4_f32(false, a1, false, bf1,
                                                (short)0, acc, false, false);

    // D layout: VGPR p, lanes 0-15 = row p, col = lane. Rows 0..5 = forces.
    // 16 consecutive dword addresses per atomic -> one L2 transaction each.
    if (!hi) {
      atomicAdd(&f_acc[off0 + utb], acc[0]);
      atomicAdd(&f_acc[off1 + utb], acc[1]);
      atomicAdd(&f_acc[off2 + utb], acc[2]);
      atomicAdd(&f_acc[off3 + utb], acc[3]);
      atomicAdd(&f_acc[off4 + utb], acc[4]);
      atomicAdd(&f_acc[off5 + utb], acc[5]);
    }
  }
}

// ------------------------------------------------------- residual + reduce
// f_acc and reaction_hist share layout [N,2,T]. T is a multiple of 4, so a
// 4-aligned float4 chunk lies within a single (node,comp) row -> one ext/mask
// fetch per 4 residuals, fully coalesced 128-bit streams.
__global__ void fem_loss_reduce(const float4* __restrict__ f_acc4,
                                const float*  __restrict__ ext,   // [N*2]
                                const float4* __restrict__ reac4, // [N*2, T/4]
                                const int*    __restrict__ dirb,  // [N*2]
                                float* __restrict__ out,
                                long total4, int tshift4) {       // tshift4 = log2(T/4)
  __shared__ float sm[256];
  float acc = 0.0f;
  long i = (long)blockIdx.x * blockDim.x + threadIdx.x;
  long stride = (long)gridDim.x * blockDim.x;
  for (; i < total4; i += stride) {
    long ncdof = i >> tshift4;            // (node,comp) flat index
    float mask = (float)dirb[ncdof];
    float ex = ext[ncdof];
    float4 f = f_acc4[i];
    float4 rc = reac4[i];
    float r0 = f.x - ex - mask * rc.x;
    float r1 = f.y - ex - mask * rc.y;
    float r2 = f.z - ex - mask * rc.z;
    float r3 = f.w - ex - mask * rc.w;
    acc += r0 * r0 + r1 * r1 + r2 * r2 + r3 * r3;
  }
  sm[threadIdx.x] = acc;
  __syncthreads();
  for (int o = 128; o > 0; o >>= 1) {
    if ((int)threadIdx.x < o) sm[threadIdx.x] += sm[threadIdx.x + o];
    __syncthreads();
  }
  if (threadIdx.x == 0) atomicAdd(out, sm[0]);
}

// -------------------------------------------------------------- launch glue
extern "C" void kernel_launch(void* const* d_in, const int* in_sizes, int n_in,
                              void* d_out, int out_size, void* d_ws, size_t ws_size,
                              hipStream_t stream) {
  const float* node_coords = (const float*)d_in[0];  // [N,2]
  const float* disps_hist  = (const float*)d_in[1];  // [T,N,2]
  const float* reac_hist   = (const float*)d_in[2];  // [N,2,T]
  const float* ext_forces  = (const float*)d_in[3];  // [N,2]
  const float* lam         = (const float*)d_in[4];  // [E]
  const float* mu          = (const float*)d_in[5];  // [E]
  const int*   conn        = (const int*)d_in[6];    // [E,3]
  const int*   dirb        = (const int*)d_in[7];    // [N,2]

  const int n_nodes = in_sizes[0] / 2;
  const int n_elem  = in_sizes[4];
  const int T       = in_sizes[1] / in_sizes[0];     // 64
  const long n2     = 2L * n_nodes;
  const long ntot   = n2 * T;                        // 12.8M floats

  // scratch: u_tr [N*2,T] then f_acc [N*2,T]  (2 x 51.2 MB)
  float* u_tr  = (float*)d_ws;
  float* f_acc = u_tr + ntot;
  float* out   = (float*)d_out;

  int tshift = 0;
  while ((1 << tshift) < T) ++tshift;                // T is a power of two (64)

  // 1) zero the force accumulator and the scalar output (every call)
  fem_zero_f32v4<<<4096, 256, 0, stream>>>((float4*)f_acc, ntot / 4);
  fem_zero_f32<<<1, 32, 0, stream>>>(out, out_size);

  // 2) transpose displacements to node-major time-contiguous layout
  {
    dim3 grid((unsigned)((n2 + 255) / 256), (unsigned)T, 1);
    fem_transpose_t_major_to_n_major<<<grid, 256, 0, stream>>>(
        disps_hist, u_tr, (int)n2, T);
  }

  // 3) element-stiffness x time-batched displacements via f32 WMMA + scatter
  const int waves_per_block = 8;                     // 256 threads, wave32
  const int eblocks = (n_elem + waves_per_block - 1) / waves_per_block;
  fem_elem_forces_wmma<<<eblocks, 256, 0, stream>>>(
      node_coords, u_tr, lam, mu, conn, f_acc, n_elem, T);

  // 4) masked residual, sum of squares (float4 streams)
  fem_loss_reduce<<<4096, 256, 0, stream>>>(
      (const float4*)f_acc, ext_forces, (const float4*)reac_hist, dirb, out,
      ntot / 4, tshift - 2);
}